// GCNConv_15899968930134
// MI455X (gfx1250) — compile-verified
//
#include <hip/hip_runtime.h>
#include <hip/hip_bf16.h>

typedef float v2f __attribute__((ext_vector_type(2)));
typedef float v8f __attribute__((ext_vector_type(8)));

#define N_NODES 100000
#define N_EDGES 640000
#define CH      128

// ---------------------------------------------------------------------------
// zero fill (grid-stride)
__global__ void zero_kernel(float* __restrict__ p, size_t n) {
    size_t i = (size_t)blockIdx.x * blockDim.x + threadIdx.x;
    size_t stride = (size_t)gridDim.x * blockDim.x;
    for (; i < n; i += stride) p[i] = 0.0f;
}

// ---------------------------------------------------------------------------
// deg[row[e]] += vals[e]
__global__ void deg_kernel(const int* __restrict__ row, const float* __restrict__ vals,
                           float* __restrict__ deg, int E) {
    int e = blockIdx.x * blockDim.x + threadIdx.x;
    if (e < E) {
        __hip_atomic_fetch_add(&deg[row[e]], vals[e],
                               __ATOMIC_RELAXED, __HIP_MEMORY_SCOPE_AGENT);
    }
}

// D = rsqrt(deg + 1), in place
__global__ void rsqrt_kernel(float* __restrict__ d, int N) {
    int i = blockIdx.x * blockDim.x + threadIdx.x;
    if (i < N) d[i] = rsqrtf(d[i] + 1.0f);
}

// ---------------------------------------------------------------------------
// Y = D[:,None] * (X @ W)   via V_WMMA_F32_16X16X4_F32
// block = 128 threads (4 waves), one 16-row M tile per block,
// wave w computes N tiles [32w, 32w+16) and [32w+16, 32w+32). K loop: 128 in steps of 4.
__global__ void __launch_bounds__(128)
gemm_scale_kernel(const float* __restrict__ X, const float* __restrict__ W,
                  const float* __restrict__ D, float* __restrict__ Y) {
    __shared__ float sX[16 * 132];           // 16 rows x 128 cols, +4 pad

    const int tid  = threadIdx.x;
    const int row0 = blockIdx.x * 16;

    // cooperative, coalesced load of the X tile (512 float4 chunks)
    for (int i = tid; i < 16 * 32; i += 128) {
        const int r  = i >> 5;               // row 0..15
        const int c4 = i & 31;               // float4 column 0..31
        const float4 v = ((const float4*)(X + (size_t)(row0 + r) * CH))[c4];
        float* dst = &sX[r * 132 + c4 * 4];
        dst[0] = v.x; dst[1] = v.y; dst[2] = v.z; dst[3] = v.w;
    }
    __syncthreads();

    const int wave = tid >> 5;
    const int lane = tid & 31;
    const int m    = lane & 15;              // M (for A) / N (for B) within tile
    const int half = lane >> 4;

    const int n0 = wave * 32;
    const int n1 = n0 + 16;

    v8f acc0 = {};
    v8f acc1 = {};

    for (int k0 = 0; k0 < CH; k0 += 4) {
        const int ka = k0 + 2 * half;        // per-lane K base
        v2f a;
        a.x = sX[m * 132 + ka];
        a.y = sX[m * 132 + ka + 1];

        v2f b0, b1;
        b0.x = W[(size_t)ka * CH + n0 + m];
        b0.y = W[(size_t)(ka + 1) * CH + n0 + m];
        b1.x = W[(size_t)ka * CH + n1 + m];
        b1.y = W[(size_t)(ka + 1) * CH + n1 + m];

        acc0 = __builtin_amdgcn_wmma_f32_16x16x4_f32(
            false, a, false, b0, (short)0, acc0, false, false);
        acc1 = __builtin_amdgcn_wmma_f32_16x16x4_f32(
            false, a, false, b1, (short)0, acc1, false, false);
    }

    // C/D layout: VGPR r -> M = r + 8*half, N = lane&15
#pragma unroll
    for (int r = 0; r < 8; ++r) {
        const int rown = row0 + r + 8 * half;
        const float d  = D[rown];
        Y[(size_t)rown * CH + n0 + m] = acc0[r] * d;
        Y[(size_t)rown * CH + n1 + m] = acc1[r] * d;
    }
}

// ---------------------------------------------------------------------------
// SpMM scatter: one wave per edge; 32 lanes x float4 = 128 channels.
// AY[row[e], :] += vals[e] * Y[col[e], :]
__global__ void __launch_bounds__(256)
spmm_kernel(const int* __restrict__ row, const int* __restrict__ col,
            const float* __restrict__ vals, const float* __restrict__ Y,
            float* __restrict__ AY, int E) {
    const int e    = (int)(((size_t)blockIdx.x * blockDim.x + threadIdx.x) >> 5);
    const int lane = threadIdx.x & 31;
    if (e >= E) return;

    const int   r = row[e];
    const int   c = col[e];
    const float v = vals[e];

    const float4 y = ((const float4*)(Y + (size_t)c * CH))[lane];
    float* dst = AY + (size_t)r * CH + lane * 4;
    __hip_atomic_fetch_add(dst + 0, v * y.x, __ATOMIC_RELAXED, __HIP_MEMORY_SCOPE_AGENT);
    __hip_atomic_fetch_add(dst + 1, v * y.y, __ATOMIC_RELAXED, __HIP_MEMORY_SCOPE_AGENT);
    __hip_atomic_fetch_add(dst + 2, v * y.z, __ATOMIC_RELAXED, __HIP_MEMORY_SCOPE_AGENT);
    __hip_atomic_fetch_add(dst + 3, v * y.w, __ATOMIC_RELAXED, __HIP_MEMORY_SCOPE_AGENT);
}

// ---------------------------------------------------------------------------
// out = D[:,None] * out + bias   (in place over d_out)
__global__ void finalize_kernel(float* __restrict__ out, const float* __restrict__ D,
                                const float* __restrict__ bias, size_t total) {
    size_t i = (size_t)blockIdx.x * blockDim.x + threadIdx.x;
    size_t stride = (size_t)gridDim.x * blockDim.x;
    for (; i < total; i += stride) {
        const size_t r = i >> 7;       // /128
        const int    c = (int)(i & 127);
        out[i] = D[r] * out[i] + bias[c];
    }
}

// ---------------------------------------------------------------------------
extern "C" void kernel_launch(void* const* d_in, const int* in_sizes, int n_in,
                              void* d_out, int out_size, void* d_ws, size_t ws_size,
                              hipStream_t stream) {
    const int*   row  = (const int*)d_in[0];
    const int*   col  = (const int*)d_in[1];
    const float* vals = (const float*)d_in[2];
    const float* X    = (const float*)d_in[3];
    const float* W    = (const float*)d_in[4];
    const float* bias = (const float*)d_in[5];
    float* out = (float*)d_out;

    // workspace layout: [0, N) deg->D ; [Y_OFF, Y_OFF + N*CH) Y
    float* D = (float*)d_ws;
    const size_t Y_OFF = ((size_t)N_NODES + 511) & ~(size_t)511;  // 100352, 16B aligned
    float* Y = (float*)d_ws + Y_OFF;

    const size_t total = (size_t)N_NODES * CH;

    // 1) deg = segment_sum(vals, row);  D = rsqrt(deg + 1)
    zero_kernel<<<(N_NODES + 255) / 256, 256, 0, stream>>>(D, (size_t)N_NODES);
    deg_kernel<<<(N_EDGES + 255) / 256, 256, 0, stream>>>(row, vals, D, N_EDGES);
    rsqrt_kernel<<<(N_NODES + 255) / 256, 256, 0, stream>>>(D, N_NODES);

    // 2) zero the AY accumulator (d_out)
    zero_kernel<<<2048, 256, 0, stream>>>(out, total);

    // 3) Y = D * (X @ W) via WMMA f32 16x16x4
    gemm_scale_kernel<<<N_NODES / 16, 128, 0, stream>>>(X, W, D, Y);

    // 4) AY += vals * Y[col]  (one wave per edge)
    {
        const size_t threads = (size_t)N_EDGES * 32;
        const int blocks = (int)((threads + 255) / 256);
        spmm_kernel<<<blocks, 256, 0, stream>>>(row, col, vals, Y, out, N_EDGES);
    }

    // 5) out = D * AY + bias
    finalize_kernel<<<2048, 256, 0, stream>>>(out, D, bias, total);
}